// MoETransformerDecoderBlock_81930796138559
// MI455X (gfx1250) — compile-verified
//
#include <hip/hip_runtime.h>

typedef float v2f __attribute__((ext_vector_type(2)));
typedef float v8f __attribute__((ext_vector_type(8)));

#define WMMA_F32_K4(A_, B_, C_) \
  __builtin_amdgcn_wmma_f32_16x16x4_f32(false, (A_), false, (B_), (short)0, (C_), false, false)

// Problem constants (from reference: B=4, S=2048, D=1024, H=16, E=8, top-2)
constexpr int Bc   = 4;
constexpr int Sc   = 2048;
constexpr int Dc   = 1024;
constexpr int Hc   = 16;
constexpr int Dh   = 64;
constexpr int Ec   = 8;
constexpr int Mtok = Bc * Sc;   // 8192 tokens

__device__ __forceinline__ v8f zero8() {
  v8f z = {0.f, 0.f, 0.f, 0.f, 0.f, 0.f, 0.f, 0.f};
  return z;
}

// ---------------------------------------------------------------------------
// Tiled WMMA-f32 GEMM: out = A[8192,1024] * Bw[1024,1024] + bias
// Block tile 128(M) x 128(N), BK=16, 8 waves, each wave a 32x64 subtile
// (2x4 fragments: 8 WMMAs per kk-step off 2 A-frags + 4 B-frags).
// EPI: 0 = head layout [B,H,S,Dh] (Q and K), 1 = transposed [B,H,Dh,S] (V),
//      3 = plain row-major [M,N]
// ---------------------------------------------------------------------------
template <int EPI>
__global__ __launch_bounds__(256) void gemm_wmma_k(const float* __restrict__ A,
                                                   const float* __restrict__ Bw,
                                                   const float* __restrict__ bias,
                                                   float* __restrict__ out) {
  constexpr int LDA = 20;   // padded: conflict-free A fragment reads
  constexpr int LDB = 136;  // padded: 2*LDB == 16 (mod 64) -> disjoint halves
  __shared__ float sA[128 * LDA];
  __shared__ float sB[16 * LDB];

  const int tid  = threadIdx.x;
  const int w    = tid >> 5;
  const int lane = tid & 31;
  const int l16  = lane & 15;
  const int hs   = lane >> 4;       // half-wave select (K interleave)
  const int wm   = w & 3;           // wave M group (32 rows each)
  const int wn   = w >> 2;          // wave N group (64 cols each)
  const int rowBase = blockIdx.y * 128;
  const int colBase = blockIdx.x * 128;

  v8f acc[2][4];
#pragma unroll
  for (int i = 0; i < 2; ++i)
#pragma unroll
    for (int j = 0; j < 4; ++j) acc[i][j] = zero8();

  for (int k0 = 0; k0 < Dc; k0 += 16) {
    __syncthreads();
    // A tile 128x16 (2 float4 per thread)
#pragma unroll
    for (int i = 0; i < 2; ++i) {
      int f = tid + i * 256;
      int r = f >> 2, c4 = (f & 3) * 4;
      float4 va = *(const float4*)(A + (size_t)(rowBase + r) * Dc + k0 + c4);
      float* d = sA + r * LDA + c4;
      d[0] = va.x; d[1] = va.y; d[2] = va.z; d[3] = va.w;
    }
    // B tile 16x128 (2 float4 per thread)
#pragma unroll
    for (int i = 0; i < 2; ++i) {
      int f = tid + i * 256;
      int r = f >> 5, c4 = (f & 31) * 4;
      *(float4*)(sB + r * LDB + c4) =
          *(const float4*)(Bw + (size_t)(k0 + r) * Dc + colBase + c4);
    }
    __syncthreads();

#pragma unroll
    for (int kk = 0; kk < 16; kk += 4) {
      int kb = kk + hs * 2;
      v2f a0 = *(const v2f*)(sA + (wm * 32 + l16) * LDA + kb);
      v2f a1 = *(const v2f*)(sA + (wm * 32 + 16 + l16) * LDA + kb);
      v2f b[4];
#pragma unroll
      for (int j = 0; j < 4; ++j) {
        b[j].x = sB[kb * LDB + wn * 64 + j * 16 + l16];
        b[j].y = sB[(kb + 1) * LDB + wn * 64 + j * 16 + l16];
      }
#pragma unroll
      for (int j = 0; j < 4; ++j) {
        acc[0][j] = WMMA_F32_K4(a0, b[j], acc[0][j]);
        acc[1][j] = WMMA_F32_K4(a1, b[j], acc[1][j]);
      }
    }
  }

  // Epilogue. D-frag: VGPR r -> row (r + 8*hs), col l16 within 16x16 tile.
#pragma unroll
  for (int i = 0; i < 2; ++i) {
#pragma unroll
    for (int r = 0; r < 8; ++r) {
      int m = rowBase + wm * 32 + i * 16 + hs * 8 + r;
#pragma unroll
      for (int j = 0; j < 4; ++j) {
        int n = colBase + wn * 64 + j * 16 + l16;
        float val = acc[i][j][r] + bias[n];
        if constexpr (EPI == 3) {
          out[(size_t)m * Dc + n] = val;
        } else {
          int bb = m >> 11, ss = m & (Sc - 1);
          int hh = n >> 6, dd = n & (Dh - 1);
          if constexpr (EPI == 1)
            out[((size_t)(bb * Hc + hh) * Dh + dd) * Sc + ss] = val;  // V^T
          else
            out[((size_t)(bb * Hc + hh) * Sc + ss) * Dh + dd] = val;  // Q / K
        }
      }
    }
  }
}

// ---------------------------------------------------------------------------
// Flash attention: one block = 32 query rows of one (b,h).
// Online softmax; QK^T and P.V via f32 WMMA.
// K in [B,H,S,Dh] (d fastest -> contiguous b64 B-frags for QK^T);
// V in [B,H,Dh,S] (key fastest -> contiguous b64 B-frags for P.V).
// Writes ctx in [B,S,D] layout.
// ---------------------------------------------------------------------------
__global__ __launch_bounds__(256) void attention_k(const float* __restrict__ Qb,
                                                   const float* __restrict__ Kb,
                                                   const float* __restrict__ Vb,
                                                   float* __restrict__ ctx) {
  constexpr int LQ = 68;
  constexpr int LK = 68;
  __shared__ float sQ[32 * LQ];   // [q][d]
  __shared__ float sK[64 * LK];   // [key][d]
  __shared__ float sV[64 * LK];   // [dh][key]
  __shared__ float sP[32 * LQ];   // [q][key]
  __shared__ float sRed[4 * 32];
  __shared__ float sM[32], sL[32], sAl[32];

  const int tid  = threadIdx.x;
  const int w    = tid >> 5;
  const int lane = tid & 31;
  const int l16  = lane & 15;
  const int hs   = lane >> 4;
  const int qi   = w & 1;   // 16-row group (shared by S and O tiles)
  const int kj   = w >> 1;  // 16-col group
  const int bh    = blockIdx.y;
  const int qbase = blockIdx.x * 32;

  const float* Qp = Qb + (size_t)bh * Sc * Dh;
  const float* Kp = Kb + (size_t)bh * Sc * Dh;
  const float* Vp = Vb + (size_t)bh * Dh * Sc;

  // Load Q tile 32x64
#pragma unroll
  for (int i = 0; i < 2; ++i) {
    int f = tid + i * 256;
    int r = f >> 4, c4 = (f & 15) * 4;
    *(float4*)(sQ + r * LQ + c4) =
        *(const float4*)(Qp + (size_t)(qbase + r) * Dh + c4);
  }
  if (tid < 32) { sM[tid] = -1e30f; sL[tid] = 0.f; }
  v8f oacc = zero8();
  __syncthreads();

  for (int c = 0; c < Sc / 64; ++c) {
    int kbase = c * 64;
    __syncthreads();  // previous P.V reads done before overwriting sK/sV
#pragma unroll
    for (int i = 0; i < 4; ++i) {
      int f = tid + i * 256;
      int r = f >> 4, c4 = (f & 15) * 4;
      *(float4*)(sK + r * LK + c4) =
          *(const float4*)(Kp + (size_t)(kbase + r) * Dh + c4);   // [key][d]
      *(float4*)(sV + r * LK + c4) =
          *(const float4*)(Vp + (size_t)r * Sc + kbase + c4);     // [dh][key]
    }
    __syncthreads();

    // S = (Q . K^T) * 1/sqrt(Dh)  -- wave tile 16x16
    v8f s = zero8();
#pragma unroll
    for (int kk = 0; kk < 64; kk += 4) {
      int kb = kk + hs * 2;
      v2f a = *(const v2f*)(sQ + (qi * 16 + l16) * LQ + kb);
      v2f b = *(const v2f*)(sK + (kj * 16 + l16) * LK + kb);
      s = WMMA_F32_K4(a, b, s);
    }
#pragma unroll
    for (int r = 0; r < 8; ++r) s[r] *= 0.125f;

    // Per-tile row max -> sRed
#pragma unroll
    for (int r = 0; r < 8; ++r) {
      float red = s[r];
      red = fmaxf(red, __shfl_xor(red, 1));
      red = fmaxf(red, __shfl_xor(red, 2));
      red = fmaxf(red, __shfl_xor(red, 4));
      red = fmaxf(red, __shfl_xor(red, 8));
      if (l16 == 0) sRed[kj * 32 + qi * 16 + hs * 8 + r] = red;
    }
    __syncthreads();
    if (tid < 32) {
      float mo = sM[tid];
      float cm = fmaxf(fmaxf(sRed[tid], sRed[32 + tid]),
                       fmaxf(sRed[64 + tid], sRed[96 + tid]));
      float mn  = fmaxf(mo, cm);
      sAl[tid] = __expf(mo - mn);
      sM[tid]  = mn;
    }
    __syncthreads();

    // P = exp(S - m), stage into sP, partial row sums; rescale O accum
#pragma unroll
    for (int r = 0; r < 8; ++r) {
      int row = qi * 16 + hs * 8 + r;
      float p = __expf(s[r] - sM[row]);
      sP[row * LQ + kj * 16 + l16] = p;
      float red = p;
      red += __shfl_xor(red, 1);
      red += __shfl_xor(red, 2);
      red += __shfl_xor(red, 4);
      red += __shfl_xor(red, 8);
      if (l16 == 0) sRed[kj * 32 + row] = red;
      oacc[r] *= sAl[row];
    }
    __syncthreads();
    if (tid < 32) {
      float sum = sRed[tid] + sRed[32 + tid] + sRed[64 + tid] + sRed[96 + tid];
      sL[tid] = sAl[tid] * sL[tid] + sum;
    }
    __syncthreads();

    // O += P . V  -- wave tile 16(q) x 16(dh)
#pragma unroll
    for (int kk = 0; kk < 64; kk += 4) {
      int kb = kk + hs * 2;
      v2f a = *(const v2f*)(sP + (qi * 16 + l16) * LQ + kb);
      v2f b = *(const v2f*)(sV + (kj * 16 + l16) * LK + kb);
      oacc = WMMA_F32_K4(a, b, oacc);
    }
  }
  __syncthreads();

  const int b_ = bh >> 4, h_ = bh & 15;
#pragma unroll
  for (int r = 0; r < 8; ++r) {
    int row = qi * 16 + hs * 8 + r;
    float val = oacc[r] / sL[row];
    int srow = qbase + row;
    int col  = kj * 16 + l16;
    ctx[((size_t)b_ * Sc + srow) * Dc + h_ * Dh + col] = val;
  }
}

// ---------------------------------------------------------------------------
// Residual + LayerNorm: out = resid + LN(s1 (+ s2)) * g + be
// ---------------------------------------------------------------------------
__global__ __launch_bounds__(256) void ln_residual_k(const float* __restrict__ resid,
                                                     const float* __restrict__ s1,
                                                     const float* __restrict__ s2,
                                                     const float* __restrict__ g,
                                                     const float* __restrict__ be,
                                                     float* __restrict__ out) {
  __shared__ float r1[256], r2[256];
  __shared__ float stats[2];
  const int t = blockIdx.x, tid = threadIdx.x;
  float v[4];
  float su = 0.f, sq = 0.f;
#pragma unroll
  for (int i = 0; i < 4; ++i) {
    int d = tid + i * 256;
    float x = s1[(size_t)t * Dc + d];
    if (s2) x += s2[(size_t)t * Dc + d];
    v[i] = x;
    su += x;
    sq += x * x;
  }
  r1[tid] = su; r2[tid] = sq;
  __syncthreads();
  for (int s = 128; s > 0; s >>= 1) {
    if (tid < s) { r1[tid] += r1[tid + s]; r2[tid] += r2[tid + s]; }
    __syncthreads();
  }
  if (tid == 0) {
    float mean = r1[0] * (1.f / Dc);
    float var  = r2[0] * (1.f / Dc) - mean * mean;
    stats[0] = mean;
    stats[1] = rsqrtf(var + 1e-5f);
  }
  __syncthreads();
  float mean = stats[0], rstd = stats[1];
#pragma unroll
  for (int i = 0; i < 4; ++i) {
    int d = tid + i * 256;
    out[(size_t)t * Dc + d] =
        resid[(size_t)t * Dc + d] + (v[i] - mean) * rstd * g[d] + be[d];
  }
}

// ---------------------------------------------------------------------------
// Gate: logits = x1 @ Wg + bg; softmax; top-2; bucket tokens per expert.
// ---------------------------------------------------------------------------
__global__ __launch_bounds__(256) void gate_k(const float* __restrict__ x1,
                                              const float* __restrict__ Wg,
                                              const float* __restrict__ bg,
                                              int* __restrict__ busTok,
                                              int* __restrict__ busSlot,
                                              float* __restrict__ busGate,
                                              int* __restrict__ cursor) {
  __shared__ float sred[256];
  __shared__ float logits[Ec];
  const int t = blockIdx.x, tid = threadIdx.x;
  float acc[Ec];
#pragma unroll
  for (int e = 0; e < Ec; ++e) acc[e] = 0.f;
  for (int d = tid; d < Dc; d += 256) {
    float xv = x1[(size_t)t * Dc + d];
#pragma unroll
    for (int e = 0; e < Ec; ++e) acc[e] += xv * Wg[d * Ec + e];
  }
  for (int e = 0; e < Ec; ++e) {
    sred[tid] = acc[e];
    __syncthreads();
    for (int s = 128; s > 0; s >>= 1) {
      if (tid < s) sred[tid] += sred[tid + s];
      __syncthreads();
    }
    if (tid == 0) logits[e] = sred[0] + bg[e];
    __syncthreads();
  }
  if (tid == 0) {
    float mx = logits[0];
    for (int e = 1; e < Ec; ++e) mx = fmaxf(mx, logits[e]);
    float ex[Ec];
    float Z = 0.f;
    for (int e = 0; e < Ec; ++e) { ex[e] = __expf(logits[e] - mx); Z += ex[e]; }
    float inv = 1.f / Z;
    int i0 = 0; float v0 = ex[0];
    for (int e = 1; e < Ec; ++e) if (ex[e] > v0) { v0 = ex[e]; i0 = e; }
    int i1 = -1; float v1 = -1.f;
    for (int e = 0; e < Ec; ++e)
      if (e != i0 && ex[e] > v1) { v1 = ex[e]; i1 = e; }
    int p0 = atomicAdd(&cursor[i0], 1);
    busTok[i0 * Mtok + p0]  = t;
    busSlot[i0 * Mtok + p0] = 0;
    busGate[i0 * Mtok + p0] = v0 * inv;
    int p1 = atomicAdd(&cursor[i1], 1);
    busTok[i1 * Mtok + p1]  = t;
    busSlot[i1 * Mtok + p1] = 1;
    busGate[i1 * Mtok + p1] = v1 * inv;
  }
}

// ---------------------------------------------------------------------------
// Gathered per-expert MoE GEMM: picked[slot][tok] = gate*(x1[tok] @ We[e] + bexp[e])
// grid.y = e*64 + rowTile; early-exit tiles beyond this expert's token count.
// Same 128x128 / 32x64-per-wave tiling as gemm_wmma_k.
// ---------------------------------------------------------------------------
__global__ __launch_bounds__(256) void moe_gemm_k(const float* __restrict__ x1,
                                                  const float* __restrict__ We,
                                                  const float* __restrict__ bexp,
                                                  const int* __restrict__ busTok,
                                                  const int* __restrict__ busSlot,
                                                  const float* __restrict__ busGate,
                                                  const int* __restrict__ cursor,
                                                  float* __restrict__ p0,
                                                  float* __restrict__ p1) {
  constexpr int LDA = 20;
  constexpr int LDB = 136;
  __shared__ float sA[128 * LDA];
  __shared__ float sB[16 * LDB];

  const int tid  = threadIdx.x;
  const int w    = tid >> 5;
  const int lane = tid & 31;
  const int l16  = lane & 15;
  const int hs   = lane >> 4;
  const int wm   = w & 3, wn = w >> 2;

  const int e       = blockIdx.y >> 6;
  const int rowBase = (blockIdx.y & 63) * 128;
  const int cnt     = cursor[e];          // uniform across block
  if (rowBase >= cnt) return;
  const int colBase = blockIdx.x * 128;
  const float* Bw = We + (size_t)e * Dc * Dc;

  int  tokA[2];
  bool valA[2];
#pragma unroll
  for (int i = 0; i < 2; ++i) {
    int f = tid + i * 256;
    int r = rowBase + (f >> 2);
    valA[i] = (r < cnt);
    tokA[i] = valA[i] ? busTok[e * Mtok + r] : 0;
  }

  v8f acc[2][4];
#pragma unroll
  for (int i = 0; i < 2; ++i)
#pragma unroll
    for (int j = 0; j < 4; ++j) acc[i][j] = zero8();

  for (int k0 = 0; k0 < Dc; k0 += 16) {
    __syncthreads();
#pragma unroll
    for (int i = 0; i < 2; ++i) {
      int f = tid + i * 256;
      int r = f >> 2, c4 = (f & 3) * 4;
      float4 va = valA[i]
                      ? *(const float4*)(x1 + (size_t)tokA[i] * Dc + k0 + c4)
                      : make_float4(0.f, 0.f, 0.f, 0.f);
      float* d = sA + r * LDA + c4;
      d[0] = va.x; d[1] = va.y; d[2] = va.z; d[3] = va.w;
    }
#pragma unroll
    for (int i = 0; i < 2; ++i) {
      int f = tid + i * 256;
      int r = f >> 5, c4 = (f & 31) * 4;
      *(float4*)(sB + r * LDB + c4) =
          *(const float4*)(Bw + (size_t)(k0 + r) * Dc + colBase + c4);
    }
    __syncthreads();
#pragma unroll
    for (int kk = 0; kk < 16; kk += 4) {
      int kb = kk + hs * 2;
      v2f a0 = *(const v2f*)(sA + (wm * 32 + l16) * LDA + kb);
      v2f a1 = *(const v2f*)(sA + (wm * 32 + 16 + l16) * LDA + kb);
      v2f b[4];
#pragma unroll
      for (int j = 0; j < 4; ++j) {
        b[j].x = sB[kb * LDB + wn * 64 + j * 16 + l16];
        b[j].y = sB[(kb + 1) * LDB + wn * 64 + j * 16 + l16];
      }
#pragma unroll
      for (int j = 0; j < 4; ++j) {
        acc[0][j] = WMMA_F32_K4(a0, b[j], acc[0][j]);
        acc[1][j] = WMMA_F32_K4(a1, b[j], acc[1][j]);
      }
    }
  }

#pragma unroll
  for (int i = 0; i < 2; ++i) {
#pragma unroll
    for (int r = 0; r < 8; ++r) {
      int rr = rowBase + wm * 32 + i * 16 + hs * 8 + r;
      if (rr < cnt) {
        int   tok  = busTok[e * Mtok + rr];
        int   slot = busSlot[e * Mtok + rr];
        float gv   = busGate[e * Mtok + rr];
        float* outp = slot ? p1 : p0;
#pragma unroll
        for (int j = 0; j < 4; ++j) {
          int n = colBase + wn * 64 + j * 16 + l16;
          outp[(size_t)tok * Dc + n] = gv * (acc[i][j][r] + bexp[e * Dc + n]);
        }
      }
    }
  }
}

__global__ void init_k(int* cursor) {
  if (threadIdx.x < Ec) cursor[threadIdx.x] = 0;
}

// ---------------------------------------------------------------------------
extern "C" void kernel_launch(void* const* d_in, const int* in_sizes, int n_in,
                              void* d_out, int out_size, void* d_ws, size_t ws_size,
                              hipStream_t stream) {
  (void)in_sizes; (void)n_in; (void)out_size; (void)ws_size;
  const float* x    = (const float*)d_in[0];
  const float* Wq   = (const float*)d_in[1];
  const float* bq   = (const float*)d_in[2];
  const float* Wk   = (const float*)d_in[3];
  const float* bk   = (const float*)d_in[4];
  const float* Wv   = (const float*)d_in[5];
  const float* bv   = (const float*)d_in[6];
  const float* Wo   = (const float*)d_in[7];
  const float* bo   = (const float*)d_in[8];
  const float* g1   = (const float*)d_in[9];
  const float* be1  = (const float*)d_in[10];
  const float* g2   = (const float*)d_in[11];
  const float* be2  = (const float*)d_in[12];
  const float* Wg   = (const float*)d_in[13];
  const float* bg   = (const float*)d_in[14];
  const float* We   = (const float*)d_in[15];
  const float* bexp = (const float*)d_in[16];
  float* out = (float*)d_out;

  float* W = (float*)d_ws;
  const size_t R = (size_t)Mtok * Dc;   // 8.39M floats per region (32 MB)
  float* Qb = W;
  float* Kb = W + R;
  float* Vb = W + 2 * R;
  float* Cb = W + 3 * R;
  // lifetime-based aliases
  float* proj = Qb;   // Q dead after attention
  float* x1   = Kb;   // K dead after attention
  float* p0   = Vb;   // V dead after attention
  float* p1   = Cb;   // ctx dead after Wo projection
  int*   busTok  = (int*)(W + 4 * R);
  int*   busSlot = busTok + Ec * Mtok;
  float* busGate = (float*)(busSlot + Ec * Mtok);
  int*   cursor  = (int*)(busGate + Ec * Mtok);

  dim3 bb(256);
  dim3 gg(Dc / 128, Mtok / 128);  // (8, 64)

  init_k<<<1, 32, 0, stream>>>(cursor);
  gemm_wmma_k<0><<<gg, bb, 0, stream>>>(x, Wq, bq, Qb);   // Q: [B,H,S,Dh]
  gemm_wmma_k<0><<<gg, bb, 0, stream>>>(x, Wk, bk, Kb);   // K: [B,H,S,Dh]
  gemm_wmma_k<1><<<gg, bb, 0, stream>>>(x, Wv, bv, Vb);   // V: [B,H,Dh,S]
  attention_k<<<dim3(Sc / 32, Bc * Hc), bb, 0, stream>>>(Qb, Kb, Vb, Cb);
  gemm_wmma_k<3><<<gg, bb, 0, stream>>>(Cb, Wo, bo, proj);
  ln_residual_k<<<Mtok, 256, 0, stream>>>(x, proj, nullptr, g1, be1, x1);
  gate_k<<<Mtok, 256, 0, stream>>>(x1, Wg, bg, busTok, busSlot, busGate, cursor);
  moe_gemm_k<<<dim3(Dc / 128, Ec * (Mtok / 128)), bb, 0, stream>>>(
      x1, We, bexp, busTok, busSlot, busGate, cursor, p0, p1);
  ln_residual_k<<<Mtok, 256, 0, stream>>>(x1, p0, p1, g2, be2, out);
}